// VolumeRaycaster_77292231459551
// MI455X (gfx1250) — compile-verified
//
#include <hip/hip_runtime.h>

#define ZIN 128
#define YIN 128
#define XIN 128
#define NCH 4
#define ZO  256
#define HO  240
#define WO  320
#define PSLICE (YIN * XIN)            // 16384 pixels per input z-slice

typedef __attribute__((ext_vector_type(2))) float v2f;
typedef __attribute__((ext_vector_type(8))) float v8f;

// 1D linear resample coefficients matching torch grid_sample
// (bilinear, padding_mode='zeros', align_corners=False) on linspace(-1,1,s_out).
__device__ __forceinline__ void lin_coeff(int oi, int s_out, int s_in,
                                          int& i0, float& w0, float& w1) {
  float t  = -1.f + 2.f * (float)oi / (float)(s_out - 1);
  float ix = ((t + 1.f) * (float)s_in - 1.f) * 0.5f;
  float f  = floorf(ix);
  i0 = (int)f;            // may be -1 (left edge) .. s_in-1; i0+1 may be s_in
  w1 = ix - f;
  w0 = 1.f - w1;
}

// -------------------------------------------------------------------------
// Pass 1: z-resample (128 -> 256) as a banded GEMM per channel, fp32 WMMA.
//   zbuf[zo][y][x][c] = sum_zi Wz[zo][zi] * vol[c][zi][y*128+x]
// One wave per 16(zo) x 16(p) C-tile. Wz is banded (2 taps/row), so a 16-row
// zo-tile only touches ~11 consecutive zi values: iterate only the 3-4
// K-chunks covering that band (skipped A entries are exact zeros ->
// identical result). Tile id and K-band bounds are forced into SGPRs via
// readfirstlane so the K loop is a scalar (s_cmp/s_cbranch) loop: EXEC stays
// all-ones around v_wmma as the ISA requires, and no per-lane loop overhead.
// A is synthesized in registers; out-of-range taps (zi=-1/128) never match
// a K index -> zeros padding for free. Output stored channel-interleaved so
// pass 2 can use b128 loads.
// -------------------------------------------------------------------------
__global__ __launch_bounds__(256) void zresample_wmma(const float* __restrict__ vol,
                                                      float* __restrict__ zbuf) {
  const int wave = __builtin_amdgcn_readfirstlane(
      blockIdx.x * (blockDim.x >> 5) + (threadIdx.x >> 5));
  const int lane = threadIdx.x & 31;
  const int half = lane >> 4;      // 0: lanes 0-15, 1: lanes 16-31
  const int ln   = lane & 15;

  // tile decode: 1024 pixel-tiles x 16 zo-tiles x 4 channels = 65536 waves
  const int pt = wave & 1023;
  const int zt = (wave >> 10) & 15;
  const int c  = wave >> 14;

  const int zo0 = zt * 16;
  const int p0  = pt * 16;

  // Per-lane output row zo = zo0 + ln (A 16x4: lanes 0-15 and 16-31 both
  // hold M = 0..15, differing only in K).
  int zi0; float w0, w1;
  lin_coeff(zo0 + ln, ZO, ZIN, zi0, w0, w1);
  const int zi1 = zi0 + 1;

  // K band covered by this zo-tile: [zi0(zo0), zi0(zo0+15)+1]  (wave-uniform)
  int band_lo, band_hi;
  {
    float d0, d1;
    lin_coeff(zo0, ZO, ZIN, band_lo, d0, d1);
    lin_coeff(zo0 + 15, ZO, ZIN, band_hi, d0, d1);
  }
  const int k_start = __builtin_amdgcn_readfirstlane(
      (band_lo < 0 ? 0 : band_lo) & ~3);
  const int k_last = __builtin_amdgcn_readfirstlane(
      band_hi + 1 > ZIN - 1 ? ZIN - 1 : band_hi + 1);

  const float* __restrict__ base = vol + (size_t)c * ZIN * PSLICE;

  v8f acc = {};
  for (int k0 = k_start; k0 <= k_last; k0 += 4) {
    // K indices handled by this lane: VGPR0 -> K = k0 + (half?2:0),
    //                                 VGPR1 -> K = k0 + (half?3:1)
    const int ka = k0 + (half ? 2 : 0);
    const int kb = ka + 1;

    v2f a, b;
    a.x = (ka == zi0) ? w0 : (ka == zi1) ? w1 : 0.f;
    a.y = (kb == zi0) ? w0 : (kb == zi1) ? w1 : 0.f;

    // B (4x16): row K striped across lanes; same K-selection pattern as A.
    b.x = base[(size_t)ka * PSLICE + p0 + ln];
    b.y = base[(size_t)kb * PSLICE + p0 + ln];

    acc = __builtin_amdgcn_wmma_f32_16x16x4_f32(false, a, false, b,
                                                (short)0, acc, false, false);
  }

  // C/D 16x16 layout: VGPR r holds row r (lanes 0-15) and row r+8 (lanes
  // 16-31). Store channel-interleaved: idx = (zo*PSLICE + p)*4 + c.
  float* __restrict__ ob =
      zbuf + ((size_t)zo0 * PSLICE + p0 + ln) * NCH + c;
#pragma unroll
  for (int r = 0; r < 8; ++r) {
    ob[(size_t)(r + half * 8) * PSLICE * NCH] = acc[r];
  }
}

// -------------------------------------------------------------------------
// Pass 2: per-pixel ray compositing. x/y bilinear taps are fixed per pixel
// (separable regular grid); zeros padding = zero the weight of out-of-range
// taps. March 256 z samples; each of the 4 corners is one 16-byte b128 load
// (all 4 channels), cumprod/weight math in registers; zbuf is L2-resident.
// A software prefetch of the z+8 slice hides L0 misses along the serial
// (cumprod) dependence chain.
// -------------------------------------------------------------------------
__global__ __launch_bounds__(256) void composite(const float* __restrict__ zbuf,
                                                 float* __restrict__ out) {
  const int pix = blockIdx.x * blockDim.x + threadIdx.x;
  if (pix >= HO * WO) return;
  const int px = pix % WO;
  const int py = pix / WO;

  int x0; float wx0, wx1;
  lin_coeff(px, WO, XIN, x0, wx0, wx1);
  int y0; float wy0, wy1;
  lin_coeff(py, HO, YIN, y0, wy0, wy1);

  if (x0 < 0)           wx0 = 0.f;
  if (x0 + 1 > XIN - 1) wx1 = 0.f;
  if (y0 < 0)           wy0 = 0.f;
  if (y0 + 1 > YIN - 1) wy1 = 0.f;
  const int xi0 = x0 < 0 ? 0 : x0;
  const int xi1 = (x0 + 1 > XIN - 1) ? (XIN - 1) : (x0 + 1);
  const int yi0 = y0 < 0 ? 0 : y0;
  const int yi1 = (y0 + 1 > YIN - 1) ? (YIN - 1) : (y0 + 1);

  const float w00 = wy0 * wx0, w01 = wy0 * wx1;
  const float w10 = wy1 * wx0, w11 = wy1 * wx1;
  const size_t o00 = (size_t)yi0 * XIN + xi0;
  const size_t o01 = (size_t)yi0 * XIN + xi1;
  const size_t o10 = (size_t)yi1 * XIN + xi0;
  const size_t o11 = (size_t)yi1 * XIN + xi1;

  const float4* __restrict__ z4 = (const float4*)zbuf;  // [zo][p] -> 4ch
  const float dscale = 100.0f / 256.0f;  // DENSITY_FACTOR / RAY_SAMPLES

  float T = 1.f, wsum = 0.f, ar = 0.f, ag = 0.f, ab = 0.f;
  for (int zo = 0; zo < ZO; ++zo) {
    const size_t zb = (size_t)zo * PSLICE;
    // prefetch 8 slices ahead (wraps at the end; stays inside the buffer)
    __builtin_prefetch(z4 + ((size_t)((zo + 8) & (ZO - 1)) * PSLICE + o00), 0, 3);
    const float4 v00 = z4[zb + o00];
    const float4 v01 = z4[zb + o01];
    const float4 v10 = z4[zb + o10];
    const float4 v11 = z4[zb + o11];
    const float cr = w00 * v00.x + w01 * v01.x + w10 * v10.x + w11 * v11.x;
    const float cg = w00 * v00.y + w01 * v01.y + w10 * v10.y + w11 * v11.y;
    const float cb = w00 * v00.z + w01 * v01.z + w10 * v10.z + w11 * v11.z;
    const float cd = w00 * v00.w + w01 * v01.w + w10 * v10.w + w11 * v11.w;

    const float d = dscale * cd;     // scaling commutes with linear interp
    T *= (1.f - d);                  // inclusive cumprod of (1-d)
    const float w = d * T;
    wsum += w;
    ar += w * cr;
    ag += w * cg;
    ab += w * cb;
  }
  const float alpha = 1.f - T;
  const float scale = alpha / (wsum + 1e-8f);
  out[0 * (HO * WO) + pix] = ar * scale;
  out[1 * (HO * WO) + pix] = ag * scale;
  out[2 * (HO * WO) + pix] = ab * scale;
  out[3 * (HO * WO) + pix] = alpha;
}

extern "C" void kernel_launch(void* const* d_in, const int* in_sizes, int n_in,
                              void* d_out, int out_size, void* d_ws, size_t ws_size,
                              hipStream_t stream) {
  (void)in_sizes; (void)n_in; (void)out_size; (void)ws_size;
  const float* vol = (const float*)d_in[0];  // (1,4,128,128,128) fp32
  float* zbuf = (float*)d_ws;                // needs 256*16384*4ch*4B = 64 MiB
  float* out  = (float*)d_out;               // (1,4,240,320) fp32

  // 65536 C-tiles, 8 waves (one tile each) per 256-thread block
  zresample_wmma<<<65536 / 8, 256, 0, stream>>>(vol, zbuf);
  composite<<<(HO * WO + 255) / 256, 256, 0, stream>>>(zbuf, out);
}